// CLIP_80144089744052
// MI455X (gfx1250) — compile-verified
//
#include <hip/hip_runtime.h>
#include <math.h>

// Problem constants (fixed by the reference).
#define B_ROWS    4096
#define DDIM      512
#define NPROTO    1000
#define NP_PAD    1024        // proto padded with zero rows -> clean tiles
#define P_STRIDE  1024
#define NCLASSES  10000
#define KSEL      10

typedef __attribute__((ext_vector_type(2))) float v2f;
typedef __attribute__((ext_vector_type(8))) float v8f;
typedef __attribute__((ext_vector_type(4))) unsigned int u32x4;
typedef __attribute__((ext_vector_type(8))) int i32x8;
typedef __attribute__((ext_vector_type(4))) int i32x4;

#if defined(__has_builtin)
#  if __has_builtin(__builtin_amdgcn_tensor_load_to_lds)
#    define HAVE_TDM 1
#  endif
#endif
#ifndef HAVE_TDM
#  define HAVE_TDM 0
#endif

__device__ __forceinline__ float wave_sum32(float s) {
#pragma unroll
  for (int m = 16; m >= 1; m >>= 1) s += __shfl_xor(s, m, 32);
  return s;
}

// ---------------------------------------------------------------------------
// Kernel 1: row-normalize text features (10000 x 512). One wave per row.
// ---------------------------------------------------------------------------
__global__ void text_norm_kernel(const float* __restrict__ text,
                                 float* __restrict__ out) {
  int wave = (blockIdx.x * blockDim.x + threadIdx.x) >> 5;
  int lane = threadIdx.x & 31;
  if (wave >= NCLASSES) return;
  const float4* src = (const float4*)(text + (size_t)wave * DDIM);
  float4 v[4];
  float s = 0.f;
#pragma unroll
  for (int i = 0; i < 4; ++i) {
    v[i] = src[lane + 32 * i];
    s += v[i].x * v[i].x + v[i].y * v[i].y + v[i].z * v[i].z + v[i].w * v[i].w;
  }
  s = wave_sum32(s);
  float inv = rsqrtf(s);
  float4* dst = (float4*)(out + (size_t)wave * DDIM);
#pragma unroll
  for (int i = 0; i < 4; ++i) {
    float4 o = v[i];
    o.x *= inv; o.y *= inv; o.z *= inv; o.w *= inv;
    dst[lane + 32 * i] = o;
  }
}

// ---------------------------------------------------------------------------
// Kernel 2: copy proto into a 1024-row zero-padded buffer (removes all tail
// masking from the WMMA inner loop). One wave per row.
// ---------------------------------------------------------------------------
__global__ void pad_proto_kernel(const float* __restrict__ proto,
                                 float* __restrict__ protoPad) {
  int row  = (blockIdx.x * blockDim.x + threadIdx.x) >> 5;
  int lane = threadIdx.x & 31;
  if (row >= NP_PAD) return;
  float4* dst = (float4*)(protoPad + (size_t)row * DDIM);
  if (row < NPROTO) {
    const float4* src = (const float4*)(proto + (size_t)row * DDIM);
#pragma unroll
    for (int i = 0; i < 4; ++i) dst[lane + 32 * i] = src[lane + 32 * i];
  } else {
    float4 z = {0.f, 0.f, 0.f, 0.f};
#pragma unroll
    for (int i = 0; i < 4; ++i) dst[lane + 32 * i] = z;
  }
}

// ---------------------------------------------------------------------------
// Kernel 3: P = img @ protoPad^T via V_WMMA_F32_16X16X4_F32.
// One wave computes a 64x32 output tile: 4 A-frags x 2 B-frags -> 8 wmmas
// per K-step. Accumulators = 8 x v8f = 64 VGPRs; total ~90 VGPRs, so NO
// scratch spills even at a 128-VGPR budget (the 64x64 version spilled).
// __launch_bounds__(32,1) tells the allocator one wave/EU is fine.
// 2048 waves total (~2 per SIMD); A/B traffic ~390 MB, all L2-resident
// (img 8 MB + proto 2 MB << 192 MB L2).
// ---------------------------------------------------------------------------
__global__ void __launch_bounds__(32, 1)
proto_gemm_kernel(const float* __restrict__ img,
                  const float* __restrict__ protoPad,
                  float* __restrict__ P) {
  int lane = threadIdx.x;
  int r    = lane & 15;
  int half = lane >> 4;
  int m0 = blockIdx.y * 64;   // 64 rows  (4 wmma tiles)
  int n0 = blockIdx.x * 32;   // 32 cols  (2 wmma tiles)

  const float* aBase = img      + (size_t)(m0 + r) * DDIM + 2 * half;
  const float* bBase = protoPad + (size_t)(n0 + r) * DDIM + 2 * half;

  v8f acc[4][2];
#pragma unroll
  for (int i = 0; i < 4; ++i)
#pragma unroll
    for (int j = 0; j < 2; ++j)
      acc[i][j] = (v8f){0.f, 0.f, 0.f, 0.f, 0.f, 0.f, 0.f, 0.f};

#pragma unroll 2
  for (int k = 0; k < DDIM; k += 4) {
    v2f a[4], b[2];
#pragma unroll
    for (int i = 0; i < 4; ++i)
      a[i] = *(const v2f*)(aBase + (size_t)i * 16 * DDIM + k);
#pragma unroll
    for (int j = 0; j < 2; ++j)
      b[j] = *(const v2f*)(bBase + (size_t)j * 16 * DDIM + k);
#pragma unroll
    for (int i = 0; i < 4; ++i)
#pragma unroll
      for (int j = 0; j < 2; ++j)
        acc[i][j] = __builtin_amdgcn_wmma_f32_16x16x4_f32(
            false, a[i], false, b[j], (short)0, acc[i][j], false, false);
  }

  // C layout: VGPR g of tile (i,j) holds (M = m0+16i+g+8*half, N = n0+16j+r)
#pragma unroll
  for (int i = 0; i < 4; ++i) {
#pragma unroll
    for (int j = 0; j < 2; ++j) {
      int col = n0 + 16 * j + r;
      bool ok = (col < NPROTO);
#pragma unroll
      for (int g = 0; g < 8; ++g) {
        int m = m0 + 16 * i + g + 8 * half;
        P[(size_t)m * P_STRIDE + col] = ok ? acc[i][j][g] : -3.0e38f;
      }
    }
  }
}

// ---------------------------------------------------------------------------
// Kernel 4: per-row top-10 of P (1000 entries). One wave per row; row lives in
// 32 regs/lane; 10 rounds of butterfly argmax, jax tie-break (lower index).
// ---------------------------------------------------------------------------
__global__ void topk_kernel(const float* __restrict__ P,
                            int* __restrict__ topk) {
  int row  = blockIdx.x * (blockDim.x >> 5) + (threadIdx.x >> 5);
  int lane = threadIdx.x & 31;
  if (row >= B_ROWS) return;
  const float* prow = P + (size_t)row * P_STRIDE;
  float v[32];
#pragma unroll
  for (int j = 0; j < 32; ++j) {
    int idx = lane + 32 * j;
    v[j] = (idx < NPROTO) ? prow[idx] : -3.4e38f;
  }
  for (int t = 0; t < KSEL; ++t) {
    float bv = -3.4e38f;
    int   bi = 0x7fffffff;
#pragma unroll
    for (int j = 0; j < 32; ++j) {
      int idx = lane + 32 * j;
      if (v[j] > bv || (v[j] == bv && idx < bi)) { bv = v[j]; bi = idx; }
    }
#pragma unroll
    for (int m = 16; m >= 1; m >>= 1) {
      float ov = __shfl_xor(bv, m, 32);
      int   oi = __shfl_xor(bi, m, 32);
      if (ov > bv || (ov == bv && oi < bi)) { bv = ov; bi = oi; }
    }
#pragma unroll
    for (int j = 0; j < 32; ++j)
      if (lane + 32 * j == bi) v[j] = -3.4e38f;   // knock winner out, no scratch
    if (lane == 0) topk[row * KSEL + t] = bi;
  }
}

// ---------------------------------------------------------------------------
// Kernel 5: final output. One 256-thread block per row. Image row is staged
// into LDS by the Tensor Data Mover (tensor_load_to_lds + s_wait_tensorcnt);
// only the 100 selected dot products are computed, then the full 10000-float
// row streams out with float4 stores -> output-bandwidth bound.
// ---------------------------------------------------------------------------
__global__ void output_kernel(const float* __restrict__ img,
                              const float* __restrict__ textn,
                              const int* __restrict__ topk,
                              const float* __restrict__ logit_scale,
                              float* __restrict__ out) {
  __shared__ float s_img[DDIM];
  __shared__ float s_val[KSEL * 10];
  __shared__ short s_slot[NPROTO];
  __shared__ int   s_idx[KSEL];

  int row = blockIdx.x;
  int tid = threadIdx.x;
  int wid = tid >> 5, lane = tid & 31;

  if (tid < KSEL) s_idx[tid] = topk[row * KSEL + tid];

#if HAVE_TDM
  if (tid < 32) {  // wave 0 drives the TDM; EXEC is ignored by tensor ops
    unsigned long long ga =
        (unsigned long long)(uintptr_t)(img + (size_t)row * DDIM);
    unsigned int lds_off = (unsigned int)(uintptr_t)(&s_img[0]); // addr[31:0] = LDS offset
    // D# group0: count=1 user descriptor, lds_addr, global_addr[56:0], type=2
    u32x4 g0;
    g0.x = 1u;
    g0.y = lds_off;
    g0.z = (unsigned int)(ga & 0xffffffffu);
    g0.w = ((unsigned int)(ga >> 32) & 0x01ffffffu) | (2u << 30);
    // D# group1: data_size=4B, tensor 512x1, tile 512x1, dim0 stride 512
    i32x8 g1;
    g1[0] = (2 << 16);          // data_size = 4 bytes
    g1[1] = (int)(512u << 16);  // tensor_dim0[15:0] = 512 (bits 63:48)
    g1[2] = (int)(1u << 16);    // tensor_dim0 hi = 0 | tensor_dim1 = 1
    g1[3] = (int)(512u << 16);  // tensor_dim1 hi = 0 | tile_dim0 = 512
    g1[4] = 1;                  // tile_dim1 = 1, tile_dim2 = 0
    g1[5] = 512;                // tensor_dim0_stride = 512 elements
    g1[6] = 0;
    g1[7] = 0;
    i32x4 z4 = {0, 0, 0, 0};
#if defined(__clang_major__) && (__clang_major__ >= 23)
    i32x8 z8 = {0, 0, 0, 0, 0, 0, 0, 0};
    __builtin_amdgcn_tensor_load_to_lds(g0, g1, z4, z4, z8, 0);
#else
    __builtin_amdgcn_tensor_load_to_lds(g0, g1, z4, z4, 0);
#endif
    __builtin_amdgcn_s_wait_tensorcnt(0);
  }
#else
  ((float2*)s_img)[tid] = ((const float2*)(img + (size_t)row * DDIM))[tid];
#endif

  for (int i = tid; i < NPROTO; i += 256) s_slot[i] = -1;
  __syncthreads();
  if (tid < KSEL) s_slot[s_idx[tid]] = (short)tid;
  __syncthreads();

  float scale = expf(logit_scale[0]);

  // 100 dot products spread over 8 waves, lane-parallel over D.
  for (int j = wid; j < KSEL * 10; j += 8) {
    int c = s_idx[j / 10] * 10 + (j % 10);
    int jn = j + 8;
    if (jn < KSEL * 10) {  // global_prefetch_b8 the next text row
      int cn = s_idx[jn / 10] * 10 + (jn % 10);
      __builtin_prefetch(textn + (size_t)cn * DDIM + lane * 16, 0, 1);
    }
    const float* trow = textn + (size_t)c * DDIM;
    float s = 0.f;
#pragma unroll
    for (int t = 0; t < 16; ++t)
      s += s_img[lane + 32 * t] * trow[lane + 32 * t];
    s = wave_sum32(s);
    if (lane == 0) s_val[j] = scale * s;
  }
  __syncthreads();

  // Stream the row: zeros except the selected class blocks.
  float4* orow = (float4*)(out + (size_t)row * NCLASSES);
  for (int f = tid; f < NCLASSES / 4; f += 256) {
    int c0 = f * 4;
    float4 o;
    float* op = (float*)&o;
#pragma unroll
    for (int e = 0; e < 4; ++e) {
      int c = c0 + e;
      int slot = s_slot[c / 10];
      op[e] = (slot < 0) ? 0.f : s_val[slot * 10 + (c % 10)];
    }
    orow[f] = o;
  }
}

// ---------------------------------------------------------------------------
extern "C" void kernel_launch(void* const* d_in, const int* in_sizes, int n_in,
                              void* d_out, int out_size, void* d_ws, size_t ws_size,
                              hipStream_t stream) {
  const float* img   = (const float*)d_in[0];  // (4096,512) pre-normalized
  const float* proto = (const float*)d_in[1];  // (1000,512) pre-normalized
  const float* text  = (const float*)d_in[2];  // (10000,512) raw
  const float* ls    = (const float*)d_in[3];  // scalar
  // d_in[4] = num_test (always 10 for this problem size)
  float* out = (float*)d_out;

  char* ws = (char*)d_ws;
  size_t off = 0;
  float* textn    = (float*)(ws + off); off += (size_t)NCLASSES * DDIM * 4;  // 20.48 MB
  float* protoPad = (float*)(ws + off); off += (size_t)NP_PAD * DDIM * 4;    //  2.10 MB
  float* P        = (float*)(ws + off); off += (size_t)B_ROWS * P_STRIDE * 4;// 16.78 MB
  int*   tk       = (int*)(ws + off);                                        //  0.16 MB

  text_norm_kernel<<<dim3((NCLASSES + 7) / 8), 256, 0, stream>>>(text, textn);
  pad_proto_kernel<<<dim3(NP_PAD / 8), 256, 0, stream>>>(proto, protoPad);
  proto_gemm_kernel<<<dim3(NP_PAD / 32, B_ROWS / 64), 32, 0, stream>>>(img, protoPad, P);
  topk_kernel<<<dim3(B_ROWS / 8), 256, 0, stream>>>(P, tk);
  output_kernel<<<dim3(B_ROWS), 256, 0, stream>>>(img, textn, tk, ls, out);
}